// HyperGraphLayer_20126216749524
// MI455X (gfx1250) — compile-verified
//
#include <hip/hip_runtime.h>
#include <hip/hip_bf16.h>

typedef float v2f __attribute__((ext_vector_type(2)));
typedef float v8f __attribute__((ext_vector_type(8)));

#define FD 64   // feature dim

__device__ __forceinline__ float sigf(float v) {
    return 1.0f / (1.0f + __expf(-v));
}

// ---------------------------------------------------------------------------
// GEMM: Y[R,64] = epilogue( X[R,64] @ W[64,64] )
// MODE 0: Y = X@W
// MODE 1: Y = sigmoid(X@W + b)
// MODE 2: Y = tanh((X@W + b) * extra)
// MODE 3: Y = tanh(X@W + b + extra)
// MODE 4: Y = sigmoid(X@W)
// One wave per 16-row tile; f32 WMMA 16x16x4, K=64 in 16 steps, 4 N-tiles.
// R must be a multiple of 16 (500000 and 200000 both are).
// ---------------------------------------------------------------------------
template <int MODE>
__global__ __launch_bounds__(256) void gemm64_kernel(
    const float* __restrict__ X, const float* __restrict__ W,
    const float* __restrict__ bias, const float* __restrict__ extra,
    float* __restrict__ Y, int R)
{
    __shared__ float sW[FD * FD];        // 16 KB, shared by all 8 waves
    __shared__ float sX[8][16 * FD];     // 4 KB per-wave X tile

    const int lane = threadIdx.x & 31;
    const int wave = threadIdx.x >> 5;

    // cooperative load of W into LDS (1024 float4 across 256 threads)
    {
        const float4* Wv  = (const float4*)W;
        float4*       sWv = (float4*)sW;
        for (int i = threadIdx.x; i < (FD * FD) / 4; i += 256) sWv[i] = Wv[i];
    }
    __syncthreads();

    const int mrow  = lane & 15;
    const int koff  = (lane < 16) ? 0 : 2;   // A/B lane K offset
    const int roff  = (lane < 16) ? 0 : 8;   // C/D lane row offset
    const int nTiles = R >> 4;

    for (int tile = blockIdx.x * 8 + wave; tile < nTiles; tile += gridDim.x * 8) {
        const int row0 = tile << 4;

        // stage this wave's 16x64 X tile into LDS (coalesced float4)
        {
            const float4* Xv  = (const float4*)(X + (size_t)row0 * FD);
            float4*       sXv = (float4*)sX[wave];
            #pragma unroll
            for (int i = 0; i < 8; ++i) sXv[lane + 32 * i] = Xv[lane + 32 * i];
        }

        v8f acc[4];
        #pragma unroll
        for (int nt = 0; nt < 4; ++nt)
            #pragma unroll
            for (int r = 0; r < 8; ++r) acc[nt][r] = 0.0f;

        const float* sXrow = &sX[wave][mrow * FD];

        #pragma unroll
        for (int k0 = 0; k0 < FD; k0 += 4) {
            v2f a;
            a.x = sXrow[k0 + koff];
            a.y = sXrow[k0 + koff + 1];
            #pragma unroll
            for (int nt = 0; nt < 4; ++nt) {
                v2f b;
                b.x = sW[(k0 + koff)     * FD + nt * 16 + mrow];
                b.y = sW[(k0 + koff + 1) * FD + nt * 16 + mrow];
                acc[nt] = __builtin_amdgcn_wmma_f32_16x16x4_f32(
                    /*neg_a=*/false, a, /*neg_b=*/false, b,
                    /*c_mod=*/(short)0, acc[nt],
                    /*reuse_a=*/false, /*reuse_b=*/false);
            }
        }

        // epilogue + store (C/D layout: VGPR r -> row r + roff, col = lane%16)
        #pragma unroll
        for (int nt = 0; nt < 4; ++nt) {
            const int col = nt * 16 + mrow;
            float bv = 0.0f;
            if (MODE == 1 || MODE == 2 || MODE == 3) bv = bias[col];
            #pragma unroll
            for (int r = 0; r < 8; ++r) {
                const int row = row0 + r + roff;
                float v = acc[nt][r] + bv;
                if (MODE == 1 || MODE == 4) {
                    v = sigf(v);
                } else if (MODE == 2) {
                    v = tanhf(v * extra[(size_t)row * FD + col]);
                } else if (MODE == 3) {
                    v = tanhf(v + extra[(size_t)row * FD + col]);
                }
                Y[(size_t)row * FD + col] = v;
            }
        }
    }
}

// ---------------------------------------------------------------------------
// zero a float buffer (float4 vectorized)
// ---------------------------------------------------------------------------
__global__ __launch_bounds__(256) void zero_kernel(float4* __restrict__ p, int n4)
{
    int i = blockIdx.x * 256 + threadIdx.x;
    if (i < n4) p[i] = make_float4(0.f, 0.f, 0.f, 0.f);
}

// ---------------------------------------------------------------------------
// node_weight[inc_node[e]] += gate[inc_hedge[e]]   (16 threads/edge, float4)
// ---------------------------------------------------------------------------
__global__ __launch_bounds__(256) void scatter_inc_kernel(
    const float* __restrict__ gate, const int* __restrict__ inc_hedge,
    const int* __restrict__ inc_node, float* __restrict__ nw, int nEdges)
{
    int t = blockIdx.x * 256 + threadIdx.x;
    int e = t >> 4;
    if (e >= nEdges) return;
    int slot = t & 15;
    const float4 g = ((const float4*)(gate + (size_t)inc_hedge[e] * FD))[slot];
    float* dst = nw + (size_t)inc_node[e] * FD + slot * 4;
    atomicAdd(dst + 0, g.x);
    atomicAdd(dst + 1, g.y);
    atomicAdd(dst + 2, g.z);
    atomicAdd(dst + 3, g.w);
}

// ---------------------------------------------------------------------------
// agg[adj_dst[e]] += Hh[adj_src[e]] * NV[adj_node[e]]
// ---------------------------------------------------------------------------
__global__ __launch_bounds__(256) void scatter_adj_kernel(
    const float* __restrict__ Hh, const float* __restrict__ NV,
    const int* __restrict__ adj_src, const int* __restrict__ adj_node,
    const int* __restrict__ adj_dst, float* __restrict__ agg, int nEdges)
{
    int t = blockIdx.x * 256 + threadIdx.x;
    int e = t >> 4;
    if (e >= nEdges) return;
    int slot = t & 15;
    const float4 h = ((const float4*)(Hh + (size_t)adj_src[e]  * FD))[slot];
    const float4 s = ((const float4*)(NV + (size_t)adj_node[e] * FD))[slot];
    float* dst = agg + (size_t)adj_dst[e] * FD + slot * 4;
    atomicAdd(dst + 0, h.x * s.x);
    atomicAdd(dst + 1, h.y * s.y);
    atomicAdd(dst + 2, h.z * s.z);
    atomicAdd(dst + 3, h.w * s.w);
}

// ---------------------------------------------------------------------------
// launch
// ---------------------------------------------------------------------------
extern "C" void kernel_launch(void* const* d_in, const int* in_sizes, int n_in,
                              void* d_out, int out_size, void* d_ws, size_t ws_size,
                              hipStream_t stream)
{
    const float* node_f   = (const float*)d_in[0];
    const float* hedge_f  = (const float*)d_in[1];
    const int*   inc_node = (const int*)d_in[2];
    const int*   inc_hedge= (const int*)d_in[3];
    const int*   adj_dst  = (const int*)d_in[4];
    const int*   adj_src  = (const int*)d_in[5];
    const int*   adj_node = (const int*)d_in[6];
    const float* Wg = (const float*)d_in[7];
    const float* bg = (const float*)d_in[8];
    const float* Wn = (const float*)d_in[9];
    const float* bn = (const float*)d_in[10];
    const float* Wh = (const float*)d_in[11];
    const float* Wv = (const float*)d_in[12];
    const float* Ws = (const float*)d_in[13];
    const float* bh = (const float*)d_in[14];

    const int N    = in_sizes[0] / FD;   // 500000
    const int H    = in_sizes[1] / FD;   // 200000
    const int nInc = in_sizes[2];        // 2000000
    const int nAdj = in_sizes[4];        // 2000000

    float* out_node  = (float*)d_out;                 // [N,64]
    float* out_hedge = out_node + (size_t)N * FD;     // [H,64]

    // workspace carve-up (bytes): gate[H*64] | nw[N*64] | agg[H*64]
    char*  ws   = (char*)d_ws;
    float* gate = (float*)ws;                                   // H*64 f32
    float* nw   = (float*)(ws + (size_t)H * FD * 4);            // N*64 f32
    float* agg  = (float*)(ws + (size_t)(H + N) * FD * 4);      // H*64 f32
    float* Hh   = gate;   // reuse: gate dead after scatter_inc
    float* NV   = nw;     // reuse: node_weight dead after node GEMM

    const int gemmBlocksH = ((H >> 4) + 7) / 8;
    const int gemmBlocksN = ((N >> 4) + 7) / 8;
    const int zeroBlocksN = ((N * (FD / 4)) + 255) / 256;
    const int zeroBlocksH = ((H * (FD / 4)) + 255) / 256;
    const int incBlocks   = (nInc * 16 + 255) / 256;
    const int adjBlocks   = (nAdj * 16 + 255) / 256;

    // 1) hedge_gate = sigmoid(hedge_f @ Wg + bg)
    gemm64_kernel<1><<<gemmBlocksH, 256, 0, stream>>>(hedge_f, Wg, bg, nullptr, gate, H);
    // 2) node_weight = 0
    zero_kernel<<<zeroBlocksN, 256, 0, stream>>>((float4*)nw, N * (FD / 4));
    // 3) node_weight[inc_node] += gate[inc_hedge]
    scatter_inc_kernel<<<incBlocks, 256, 0, stream>>>(gate, inc_hedge, inc_node, nw, nInc);
    // 4) new_node = tanh((node_f @ Wn + bn) * node_weight)
    gemm64_kernel<2><<<gemmBlocksN, 256, 0, stream>>>(node_f, Wn, bn, nw, out_node, N);
    // 5) Hh = hedge_f @ Wh
    gemm64_kernel<0><<<gemmBlocksH, 256, 0, stream>>>(hedge_f, Wh, nullptr, nullptr, Hh, H);
    // 6) NV = sigmoid(new_node @ Wv)
    gemm64_kernel<4><<<gemmBlocksN, 256, 0, stream>>>(out_node, Wv, nullptr, nullptr, NV, N);
    // 7) agg = 0
    zero_kernel<<<zeroBlocksH, 256, 0, stream>>>((float4*)agg, H * (FD / 4));
    // 8) agg[adj_dst] += Hh[adj_src] * NV[adj_node]
    scatter_adj_kernel<<<adjBlocks, 256, 0, stream>>>(Hh, NV, adj_src, adj_node, adj_dst, agg, nAdj);
    // 9) new_hedge = tanh(hedge_f @ Ws + agg + bh)
    gemm64_kernel<3><<<gemmBlocksH, 256, 0, stream>>>(hedge_f, Ws, bh, agg, out_hedge, H);
}